// ExpertModule_85495618994904
// MI455X (gfx1250) — compile-verified
//
#include <hip/hip_runtime.h>
#include <math.h>

#define BQ 8
#define TQ 4096
#define DQ 1024
#define HQ 2048
#define EQ 4
#define NQ (BQ * TQ)          // 32768 tokens
#define MT 64                 // tokens per FFN workgroup
#define HC 64                 // H chunk per outer iteration

typedef __attribute__((ext_vector_type(2))) float v2f;
typedef __attribute__((ext_vector_type(8))) float v8f;
typedef __attribute__((ext_vector_type(4))) int   v4i;

#if __has_builtin(__builtin_amdgcn_global_load_async_to_lds_b128) && \
    __has_builtin(__builtin_amdgcn_s_wait_asynccnt)
#define USE_ASYNC_LDS 1
typedef __attribute__((address_space(1))) v4i* as1_v4i;   // global b128 payload ptr
typedef __attribute__((address_space(3))) v4i* as3_v4i;   // LDS b128 payload ptr
#endif

// ---------------------------------------------------------------- init ----
__global__ void moe_init(float* cnts, float* sums, int* counter) {
    int t = threadIdx.x;
    if (t < EQ) { cnts[t] = 0.0f; sums[t] = 0.0f; counter[t] = 0; }
}

// ---------------------------------------------------------------- gate ----
// one wave (32 lanes) per token: coalesced lane-strided dot against Wg rows.
__global__ __launch_bounds__(256) void moe_gate(
    const float* __restrict__ x, const float* __restrict__ Wg,
    const float* __restrict__ bg, int* __restrict__ idx,
    float* cnts, float* sums)
{
    const int lane = threadIdx.x & 31;
    const int wv   = threadIdx.x >> 5;
    const int n    = blockIdx.x * 8 + wv;
    if (n >= NQ) return;

    const float* xr = x + (size_t)n * DQ;
    float a0 = 0.f, a1 = 0.f, a2 = 0.f, a3 = 0.f;
    for (int d = lane; d < DQ; d += 32) {
        float xv = xr[d];
        float4 wvv = *(const float4*)(Wg + d * 4);   // E == 4, one row = float4
        a0 += xv * wvv.x; a1 += xv * wvv.y; a2 += xv * wvv.z; a3 += xv * wvv.w;
    }
    #pragma unroll
    for (int off = 16; off >= 1; off >>= 1) {
        a0 += __shfl_xor(a0, off, 32);
        a1 += __shfl_xor(a1, off, 32);
        a2 += __shfl_xor(a2, off, 32);
        a3 += __shfl_xor(a3, off, 32);
    }
    if (lane == 0) {
        float l0 = a0 + bg[0], l1 = a1 + bg[1], l2 = a2 + bg[2], l3 = a3 + bg[3];
        float m = fmaxf(fmaxf(l0, l1), fmaxf(l2, l3));
        float e0 = __expf(l0 - m), e1 = __expf(l1 - m),
              e2 = __expf(l2 - m), e3 = __expf(l3 - m);
        float s = e0 + e1 + e2 + e3;
        int best = 0; float bl = l0;                  // first-wins tie (matches top_k)
        if (l1 > bl) { bl = l1; best = 1; }
        if (l2 > bl) { bl = l2; best = 2; }
        if (l3 > bl) { bl = l3; best = 3; }
        float p = __expf(bl - m) / s;
        idx[n] = best;
        atomicAdd(&cnts[best], 1.0f);
        atomicAdd(&sums[best], p);
    }
}

// ------------------------------------------------------------ finalize ----
__global__ void moe_finalize(const float* cnts, const float* sums, float* avg,
                             int* basep, int* counter, float* aux_out) {
    if (threadIdx.x == 0) {
        int b = 0; float al = 0.f;
        for (int e = 0; e < EQ; ++e) {
            float c = cnts[e];
            avg[e] = (c > 0.f) ? (sums[e] / fmaxf(c, 1.0f)) : 0.0f;
            basep[e] = b;
            b += (int)c;
            float f = c / (float)NQ;
            al += f * f;
            counter[e] = 0;
        }
        basep[EQ] = b;
        *aux_out = al * (float)EQ;
    }
}

// ------------------------------------------------------------- scatter ----
__global__ __launch_bounds__(256) void moe_scatter(
    const int* __restrict__ idx, const float* __restrict__ avg,
    const int* __restrict__ basep, int* counter,
    int* __restrict__ perm, float* __restrict__ scl)
{
    int n = blockIdx.x * 256 + threadIdx.x;
    if (n >= NQ) return;
    int e = idx[n];
    int pos = basep[e] + atomicAdd(&counter[e], 1);
    perm[pos] = n;
    scl[n] = avg[e];
}

// ----------------------------------------------------------------- ffn ----
// Fused per-expert FFN: out[tok] = scale[tok] * (relu(x@W1+b1) @ W2 + b2).
// 64-token tile, X staged into 256KB of the 320KB WGP LDS via async-to-LDS
// (ASYNCcnt path, no VGPR round-trip); h never leaves LDS.
__global__ __launch_bounds__(256) void moe_ffn(
    const float* __restrict__ x,
    const float* __restrict__ W1, const float* __restrict__ bias1,
    const float* __restrict__ W2, const float* __restrict__ bias2,
    const int* __restrict__ basep, const int* __restrict__ perm,
    const float* __restrict__ scl, float* __restrict__ out)
{
    extern __shared__ float smem[];
    float* Xs = smem;                 // [MT][DQ]
    float* Hs = smem + MT * DQ;       // [MT][HC]

    const int e  = blockIdx.y;
    const int rb = basep[e];
    const int re = basep[e + 1];
    const int row0 = rb + blockIdx.x * MT;
    if (row0 >= re) return;
    const int mv = (re - row0 < MT) ? (re - row0) : MT;   // valid rows in tile

    const int tid  = threadIdx.x;
    const int lane = tid & 31;
    const int wv   = tid >> 5;        // wave 0..7
    const int l15  = lane & 15;
    const int hl   = lane >> 4;       // 0 or 1 (low/high half-wave)

    // ---- stage X tile: one row (4KB) per iteration, block-uniform branch ----
    for (int r = 0; r < MT; ++r) {
        float* ldst = Xs + r * DQ + tid * 4;
        if (r < mv) {
            int tok = perm[row0 + r];
            const float* gsrc = x + (size_t)tok * DQ + tid * 4;
#ifdef USE_ASYNC_LDS
            // CDNA5 GLOBAL_LOAD_ASYNC_TO_LDS_B128: memory -> LDS, no VGPR data.
            // generic->AS3 via low-32-bit truncation.
            __builtin_amdgcn_global_load_async_to_lds_b128(
                (as1_v4i)(size_t)gsrc,
                (as3_v4i)(uint32_t)(size_t)ldst,
                0, 0);
#else
            *(float4*)ldst = *(const float4*)gsrc;
#endif
        } else {
            *(float4*)ldst = make_float4(0.f, 0.f, 0.f, 0.f);  // zero-pad tail rows
        }
    }
#ifdef USE_ASYNC_LDS
    __builtin_amdgcn_s_wait_asynccnt(0);
#endif
    __syncthreads();

    const float* W1e = W1 + (size_t)e * DQ * HQ;
    const float* b1e = bias1 + e * HQ;
    const float* W2e = W2 + (size_t)e * HQ * DQ;
    const float* b2e = bias2 + e * DQ;

    v8f acc[4][8];                    // 4 M-tiles x 8 N-tiles = 64x128 per wave
    #pragma unroll
    for (int m = 0; m < 4; ++m)
        #pragma unroll
        for (int nt = 0; nt < 8; ++nt) acc[m][nt] = (v8f){};

    const int c0 = wv * 128;          // this wave's output-column strip

    for (int hc = 0; hc < HQ; hc += HC) {
        // ---- phase A: Hs = relu(Xs @ W1[:, hc:hc+HC] + b1) ----
        {
            const int ncol = (wv & 3) * 16;       // h-column tile in chunk
            const int m0   = (wv >> 2) * 32;      // row base; this wave: m0, m0+16
            v8f h0 = (v8f){};
            v8f h1 = (v8f){};
            const float* wcol = W1e + hc + ncol + l15;
            const float* xr0  = Xs + (m0 + l15) * DQ;
            const float* xr1  = Xs + (m0 + 16 + l15) * DQ;
            for (int k = 0; k < DQ; k += 4) {
                const int kb = k + 2 * hl;        // ISA: VGPR0=K0/K2, VGPR1=K1/K3
                v2f b, a0, a1;
                b.x  = wcol[(size_t)kb * HQ];
                b.y  = wcol[(size_t)(kb + 1) * HQ];
                a0.x = xr0[kb];  a0.y = xr0[kb + 1];
                a1.x = xr1[kb];  a1.y = xr1[kb + 1];
                h0 = __builtin_amdgcn_wmma_f32_16x16x4_f32(
                        false, a0, false, b, (short)0, h0, false, false);
                h1 = __builtin_amdgcn_wmma_f32_16x16x4_f32(
                        false, a1, false, b, (short)0, h1, false, false);
            }
            const float bias = b1e[hc + ncol + l15];
            #pragma unroll
            for (int r = 0; r < 8; ++r) {         // C/D: row = r + 8*hl, col = l15
                int rr = r + 8 * hl;
                Hs[(m0 + rr) * HC + ncol + l15]      = fmaxf(h0[r] + bias, 0.0f);
                Hs[(m0 + 16 + rr) * HC + ncol + l15] = fmaxf(h1[r] + bias, 0.0f);
            }
        }
        __syncthreads();

        // ---- phase B: acc += Hs @ W2[hc:hc+HC, c0:c0+128] ----
        for (int kh = 0; kh < HC; kh += 4) {
            const int kb = kh + 2 * hl;
            v2f a[4];
            #pragma unroll
            for (int m = 0; m < 4; ++m) {
                const float* hr = Hs + (m * 16 + l15) * HC;
                a[m].x = hr[kb];
                a[m].y = hr[kb + 1];
            }
            #pragma unroll
            for (int nt = 0; nt < 8; ++nt) {
                const float* wp = W2e + (size_t)(hc + kb) * DQ + c0 + nt * 16 + l15;
                v2f b;
                b.x = wp[0];
                b.y = wp[DQ];
                __builtin_prefetch(wp + 4 * (size_t)DQ, 0, 1);  // next k-slab -> L2
                #pragma unroll
                for (int m = 0; m < 4; ++m)
                    acc[m][nt] = __builtin_amdgcn_wmma_f32_16x16x4_f32(
                        false, a[m], false, b, (short)0, acc[m][nt], false, false);
            }
        }
        __syncthreads();              // Hs reused next chunk
    }

    // ---- epilogue: out[tok] = scale * (acc + b2) ----
    #pragma unroll
    for (int m = 0; m < 4; ++m) {
        #pragma unroll
        for (int r = 0; r < 8; ++r) {
            int row = m * 16 + r + 8 * hl;
            if (row < mv) {
                int tok = perm[row0 + row];
                float s = scl[tok];
                float* orow = out + (size_t)tok * DQ;
                #pragma unroll
                for (int nt = 0; nt < 8; ++nt) {
                    int col = c0 + nt * 16 + l15;
                    orow[col] = s * (acc[m][nt][r] + b2e[col]);
                }
            }
        }
    }
}

// -------------------------------------------------------------- launch ----
extern "C" void kernel_launch(void* const* d_in, const int* in_sizes, int n_in,
                              void* d_out, int out_size, void* d_ws, size_t ws_size,
                              hipStream_t stream) {
    const float* x   = (const float*)d_in[0];
    const float* Wg  = (const float*)d_in[1];
    const float* bg  = (const float*)d_in[2];
    const float* W1  = (const float*)d_in[3];
    const float* b1  = (const float*)d_in[4];
    const float* W2  = (const float*)d_in[5];
    const float* b2  = (const float*)d_in[6];
    float* out = (float*)d_out;                 // [N*D] out, [1] aux_loss

    // workspace layout
    float* wsf    = (float*)d_ws;
    float* cnts   = wsf;                        // 4
    float* sums   = wsf + 4;                    // 4
    float* avg    = wsf + 8;                    // 4
    int*   basep  = (int*)(wsf + 12);           // 5
    int*   counter= (int*)(wsf + 17);           // 4
    int*   idx    = (int*)(wsf + 32);           // N
    int*   perm   = idx + NQ;                   // N
    float* scl    = (float*)(perm + NQ);        // N

    const size_t smem = (size_t)(MT * DQ + MT * HC) * sizeof(float);  // 278528 B
    static bool attr_set = false;               // idempotent, not stream state
    if (!attr_set) {
        (void)hipFuncSetAttribute((const void*)moe_ffn,
                                  hipFuncAttributeMaxDynamicSharedMemorySize,
                                  (int)smem);
        attr_set = true;
    }

    moe_init    <<<1, 32, 0, stream>>>(cnts, sums, counter);
    moe_gate    <<<NQ / 8, 256, 0, stream>>>(x, Wg, bg, idx, cnts, sums);
    moe_finalize<<<1, 32, 0, stream>>>(cnts, sums, avg, basep, counter,
                                       out + (size_t)NQ * DQ);
    moe_scatter <<<NQ / 256, 256, 0, stream>>>(idx, avg, basep, counter, perm, scl);

    dim3 grid(NQ / MT, EQ);                     // per-expert tile grid; tail WGs exit
    moe_ffn<<<grid, 256, smem, stream>>>(x, W1, b1, W2, b2, basep, perm, scl, out);
}